// EmbLoss_contrast_26731876450779
// MI455X (gfx1250) — compile-verified
//
#include <hip/hip_runtime.h>
#include <math.h>

typedef float v2f __attribute__((ext_vector_type(2)));
typedef float v8f __attribute__((ext_vector_type(8)));

#define W_PIX 1.5f
#define W_DIS 1.0f
#define INV_TEMP (1.0f / 0.07f)

// ---------------------------------------------------------------- layernorm
__device__ __forceinline__ void ln4(const float* __restrict__ emb, int b,
                                    long N, long n, const float w[4],
                                    const float bi[4], float x[4]) {
  float e[4];
#pragma unroll
  for (int d = 0; d < 4; ++d)
    e[d] = emb[((size_t)b * 4 + d) * (size_t)N + (size_t)n];
  float mu = 0.25f * (e[0] + e[1] + e[2] + e[3]);
  float var = 0.f;
#pragma unroll
  for (int d = 0; d < 4; ++d) { float t = e[d] - mu; var = fmaf(t, t, var); }
  var *= 0.25f;
  float inv = rsqrtf(var + 1e-5f);
#pragma unroll
  for (int d = 0; d < 4; ++d) x[d] = (e[d] - mu) * inv * w[d] + bi[d];
}

// ---------------------------------------------------------------- zero
__global__ void k_zero(float* p, int n) {
  int i = blockIdx.x * blockDim.x + threadIdx.x;
  if (i < n) p[i] = 0.f;
}

// ---------------------------------------------------------------- seg stats
// stats[b][seg][10] = { sum_k[4], cnt_k, sum_t[4], cnt_t }
__global__ void k_seg_stats(const float* __restrict__ emb,
                            const int* __restrict__ instance,
                            const float* __restrict__ kern,
                            const float* __restrict__ tmask,
                            const float* __restrict__ lnw,
                            const float* __restrict__ lnb,
                            float* __restrict__ stats, int blocksPerB, long N,
                            int nseg) {
  __shared__ float ssum[64 * 10];
  int b = blockIdx.x / blocksPerB;
  long blk = blockIdx.x % blocksPerB;
  for (int i = threadIdx.x; i < nseg * 10; i += blockDim.x) ssum[i] = 0.f;
  __syncthreads();
  float w[4], bi[4];
#pragma unroll
  for (int d = 0; d < 4; ++d) { w[d] = lnw[d]; bi[d] = lnb[d]; }
  long start = blk * (long)(blockDim.x * 8);
  for (int it = 0; it < 8; ++it) {
    long n = start + (long)it * blockDim.x + threadIdx.x;
    if (n < N) {
      int ins = instance[(size_t)b * N + n];
      if (tmask[(size_t)b * N + n] <= 0.5f) ins = 0;
      if (ins > 0 && ins < nseg) {
        float x[4];
        ln4(emb, b, N, n, w, bi, x);
        int base = ins * 10;
        bool kk = kern[(size_t)b * N + n] > 0.5f;
#pragma unroll
        for (int d = 0; d < 4; ++d) atomicAdd(&ssum[base + 5 + d], x[d]);
        atomicAdd(&ssum[base + 9], 1.f);
        if (kk) {
#pragma unroll
          for (int d = 0; d < 4; ++d) atomicAdd(&ssum[base + d], x[d]);
          atomicAdd(&ssum[base + 4], 1.f);
        }
      }
    }
  }
  __syncthreads();
  for (int i = threadIdx.x; i < nseg * 10; i += blockDim.x)
    if (i >= 10)  // segment 0 never used downstream
      atomicAdd(&stats[(size_t)b * nseg * 10 + i], ssum[i]);
}

// ---------------------------------------------------------------- means
__global__ void k_means(const float* __restrict__ stats, float* __restrict__ m_k,
                        float* __restrict__ m_t, float* __restrict__ Mn, int B,
                        int nseg) {
  int i = blockIdx.x * blockDim.x + threadIdx.x;
  if (i >= B * nseg) return;
  int s = i % nseg;
  const float* st = stats + (size_t)i * 10;
  float ck = fmaxf(st[4], 1.f), ct = fmaxf(st[9], 1.f);
  float mt[4], nrm = 0.f;
#pragma unroll
  for (int d = 0; d < 4; ++d) {
    float vk = (s == 0) ? 0.f : st[d] / ck;
    float vt = (s == 0) ? 0.f : st[5 + d] / ct;
    m_k[(size_t)i * 4 + d] = vk;
    m_t[(size_t)i * 4 + d] = vt;
    mt[d] = vt;
    nrm = fmaf(vt, vt, nrm);
  }
  nrm = fmaxf(sqrtf(nrm), 1e-12f);
#pragma unroll
  for (int d = 0; d < 4; ++d) Mn[(size_t)i * 4 + d] = mt[d] / nrm;
}

// ---------------------------------------------------------------- pixel loss
__global__ void k_lpix(const float* __restrict__ emb,
                       const int* __restrict__ instance,
                       const float* __restrict__ tmask,
                       const float* __restrict__ lnw,
                       const float* __restrict__ lnb,
                       const float* __restrict__ m_k,
                       const float* __restrict__ stats, float* __restrict__ lpix,
                       int blocksPerB, long N, int nseg) {
  __shared__ float smk[64 * 4];
  __shared__ float sct[64];
  __shared__ float sred[256];
  int b = blockIdx.x / blocksPerB;
  long blk = blockIdx.x % blocksPerB;
  for (int i = threadIdx.x; i < nseg * 4; i += blockDim.x)
    smk[i] = m_k[(size_t)b * nseg * 4 + i];
  for (int i = threadIdx.x; i < nseg; i += blockDim.x)
    sct[i] = fmaxf(stats[((size_t)b * nseg + i) * 10 + 9], 1.f);
  __syncthreads();
  float w[4], bi[4];
#pragma unroll
  for (int d = 0; d < 4; ++d) { w[d] = lnw[d]; bi[d] = lnb[d]; }
  float acc = 0.f;
  long start = blk * (long)(blockDim.x * 8);
  for (int it = 0; it < 8; ++it) {
    long n = start + (long)it * blockDim.x + threadIdx.x;
    if (n < N) {
      int ins = instance[(size_t)b * N + n];
      if (tmask[(size_t)b * N + n] <= 0.5f) ins = 0;
      if (ins > 0 && ins < nseg) {
        float x[4];
        ln4(emb, b, N, n, w, bi, x);
        float dot = 0.f, nx = 0.f, nm = 0.f;
#pragma unroll
        for (int d = 0; d < 4; ++d) {
          float m = smk[ins * 4 + d];
          dot = fmaf(x[d], m, dot);
          nx = fmaf(x[d], x[d], nx);
          nm = fmaf(m, m, nm);
        }
        float c = dot / (fmaxf(sqrtf(nx), 1e-8f) * fmaxf(sqrtf(nm), 1e-8f));
        acc += (1.f - c) / sct[ins];
      }
    }
  }
  sred[threadIdx.x] = acc;
  __syncthreads();
  for (int o = 128; o > 0; o >>= 1) {
    if (threadIdx.x < o) sred[threadIdx.x] += sred[threadIdx.x + o];
    __syncthreads();
  }
  if (threadIdx.x == 0) atomicAdd(&lpix[b], sred[0]);
}

// ---------------------------------------------------------------- build An
__global__ void k_build_an(const float* __restrict__ emb,
                           const int* __restrict__ pix_idx,
                           const float* __restrict__ m_t,
                           const float* __restrict__ lnw,
                           const float* __restrict__ lnb, float* __restrict__ An,
                           int K, int P, int S, int S_pad, long N, int nseg) {
  int bk = blockIdx.x;
  int b = bk / K, k = bk % K;
  int s = threadIdx.x;
  if (s >= S_pad) return;
  float v[4] = {0.f, 0.f, 0.f, 0.f};
  if (s == 0) {
#pragma unroll
    for (int d = 0; d < 4; ++d)
      v[d] = m_t[((size_t)b * nseg + k + 1) * 4 + d];
  } else if (s < S) {
    long n = (long)pix_idx[(size_t)k * P + s];
    float w[4], bi[4];
#pragma unroll
    for (int d = 0; d < 4; ++d) { w[d] = lnw[d]; bi[d] = lnb[d]; }
    ln4(emb, b, N, n, w, bi, v);
  }
  if (s < S) {
    float nrm = 0.f;
#pragma unroll
    for (int d = 0; d < 4; ++d) nrm = fmaf(v[d], v[d], nrm);
    nrm = fmaxf(sqrtf(nrm), 1e-12f);
#pragma unroll
    for (int d = 0; d < 4; ++d) v[d] /= nrm;
  }
#pragma unroll
  for (int d = 0; d < 4; ++d) An[((size_t)bk * S_pad + s) * 4 + d] = v[d];
}

// ---------------------------------------------------------------- contrastive (WMMA f32 16x16x4)
__global__ void k_contrast(const float* __restrict__ An,
                           const float* __restrict__ Mn, float* __restrict__ lcon,
                           int K, int S, int S_pad, int nseg) {
  __shared__ float sAn[256 * 4];
  __shared__ float sMn[64 * 4];
  __shared__ float sAb[4][16 * 64];
  __shared__ float sLse[4][16];
  __shared__ float sRed[128];

  int bk = blockIdx.x;
  int b = bk / K, k = bk % K;
  const float* anb = An + (size_t)bk * S_pad * 4;
  for (int i = threadIdx.x; i < S_pad * 4; i += blockDim.x) sAn[i] = anb[i];
  for (int i = threadIdx.x; i < nseg * 4; i += blockDim.x) {
    int row = i >> 2;
    float v = Mn[((size_t)b * nseg + row) * 4 + (i & 3)];
    if (row == k + 1) v = 0.f;  // colmask: own segment column contributes 0
    sMn[i] = v;
  }
  __syncthreads();

  int wave = threadIdx.x >> 5;
  int lane = threadIdx.x & 31;
  int lrow = lane & 15;
  int hi = lane >> 4;  // 0: K=0,1 half / 1: K=2,3 half
  int ST = (S + 15) >> 4;
  int MT = (nseg + 15) >> 4;
  float acc = 0.f;

  for (int st = wave; st < ST; st += 4) {
    int s0 = st * 16;
    // A fragment: 16x4 f32 -> 2 VGPRs
    v2f afrag = {sAn[(s0 + lrow) * 4 + hi * 2 + 0],
                 sAn[(s0 + lrow) * 4 + hi * 2 + 1]};

    // ---- ab = An_tile (16x4) x Mn^T (4x nseg), tiles of 16 columns
    for (int mt = 0; mt < MT; ++mt) {
      int mr = mt * 16 + lrow;
      v2f bfrag = {0.f, 0.f};
      if (mr < nseg) {
        bfrag.x = sMn[mr * 4 + hi * 2 + 0];
        bfrag.y = sMn[mr * 4 + hi * 2 + 1];
      }
      v8f c = {0.f, 0.f, 0.f, 0.f, 0.f, 0.f, 0.f, 0.f};
      c = __builtin_amdgcn_wmma_f32_16x16x4_f32(false, afrag, false, bfrag,
                                                (short)0, c, false, false);
#pragma unroll
      for (int r = 0; r < 8; ++r)
        sAb[wave][(r + hi * 8) * 64 + mt * 16 + lrow] = c[r] * INV_TEMP;
    }
    asm volatile("s_wait_dscnt 0" ::: "memory");

    // ---- per-row logsumexp over nseg columns (lanes 0..15, one row each)
    if (lane < 16) {
      float mx = -1e30f;
      for (int m = 0; m < nseg; ++m) mx = fmaxf(mx, sAb[wave][lane * 64 + m]);
      float se = 0.f;
      for (int m = 0; m < nseg; ++m) se += __expf(sAb[wave][lane * 64 + m] - mx);
      sLse[wave][lane] = mx + __logf(se);
    }
    asm volatile("s_wait_dscnt 0" ::: "memory");

    // ---- aa tiles: An_tile (16x4) x An^T (4x16) per t-tile
    for (int tt = 0; tt < ST; ++tt) {
      int t0 = tt * 16;
      v2f bfrag = {sAn[(t0 + lrow) * 4 + hi * 2 + 0],
                   sAn[(t0 + lrow) * 4 + hi * 2 + 1]};
      v8f c = {0.f, 0.f, 0.f, 0.f, 0.f, 0.f, 0.f, 0.f};
      c = __builtin_amdgcn_wmma_f32_16x16x4_f32(false, afrag, false, bfrag,
                                                (short)0, c, false, false);
      int t = t0 + lrow;  // column index this lane holds
      if (t < S) {
#pragma unroll
        for (int r = 0; r < 8; ++r) {
          int srow = s0 + r + hi * 8;
          if (srow < S) {
            float aav = c[r] * INV_TEMP;
            float lse = sLse[wave][r + hi * 8];
            float mmax = fmaxf(aav, lse);
            float ce = mmax + __logf(__expf(aav - mmax) + __expf(lse - mmax)) - aav;
            acc += ce;
          }
        }
      }
    }
  }

  sRed[threadIdx.x] = acc;
  __syncthreads();
  if (threadIdx.x == 0) {
    float t = 0.f;
    for (int i = 0; i < 128; ++i) t += sRed[i];
    atomicAdd(&lcon[b], t / ((float)S * (float)S));
  }
}

// ---------------------------------------------------------------- final scalar
__global__ void k_final(const float* __restrict__ lpix,
                        const float* __restrict__ lcon, float* __restrict__ out,
                        int B) {
  if (threadIdx.x == 0 && blockIdx.x == 0) {
    float s = 0.f;
    for (int b = 0; b < B; ++b) s += W_PIX * lpix[b] + W_DIS * lcon[b];
    out[0] = s / (float)B;
  }
}

// ---------------------------------------------------------------- launch
extern "C" void kernel_launch(void* const* d_in, const int* in_sizes, int n_in,
                              void* d_out, int out_size, void* d_ws,
                              size_t ws_size, hipStream_t stream) {
  const float* emb = (const float*)d_in[0];
  const int* instance = (const int*)d_in[1];
  const float* kern = (const float*)d_in[2];
  const float* tmask = (const float*)d_in[3];
  const float* lnw = (const float*)d_in[5];
  const float* lnb = (const float*)d_in[6];
  const int* pix_idx = (const int*)d_in[7];

  int B = in_sizes[4] / 40;
  if (B <= 0) B = 1;
  long N = (long)in_sizes[1] / B;
  long KP = (long)in_sizes[7];
  long Pp = N - KP;
  if (Pp <= 0) Pp = 1;
  int K = (int)(KP / Pp);
  if (K < 1) K = 1;
  int P = (int)Pp;
  int nseg = K + 1;
  if (nseg > 64) nseg = 64;
  int S = P / 100 + 1;
  if (S > 256) S = 256;
  int S_pad = ((S + 15) / 16) * 16;

  float* ws = (float*)d_ws;
  float* stats = ws;                                   // B*nseg*10
  float* m_k = stats + (size_t)B * nseg * 10;          // B*nseg*4
  float* m_t = m_k + (size_t)B * nseg * 4;             // B*nseg*4
  float* Mn = m_t + (size_t)B * nseg * 4;              // B*nseg*4
  float* An = Mn + (size_t)B * nseg * 4;               // B*K*S_pad*4
  float* lpix = An + (size_t)B * K * S_pad * 4;        // B
  float* lcon = lpix + B;                              // B

  int nstat = B * nseg * 10;
  hipLaunchKernelGGL(k_zero, dim3((nstat + 255) / 256), dim3(256), 0, stream,
                     stats, nstat);
  hipLaunchKernelGGL(k_zero, dim3(1), dim3(256), 0, stream, lpix, 2 * B);

  int blocksPerB = (int)((N + 2047) / 2048);
  hipLaunchKernelGGL(k_seg_stats, dim3(B * blocksPerB), dim3(256), 0, stream,
                     emb, instance, kern, tmask, lnw, lnb, stats, blocksPerB, N,
                     nseg);
  hipLaunchKernelGGL(k_means, dim3((B * nseg + 255) / 256), dim3(256), 0,
                     stream, stats, m_k, m_t, Mn, B, nseg);
  hipLaunchKernelGGL(k_lpix, dim3(B * blocksPerB), dim3(256), 0, stream, emb,
                     instance, tmask, lnw, lnb, m_k, stats, lpix, blocksPerB, N,
                     nseg);
  hipLaunchKernelGGL(k_build_an, dim3(B * K), dim3(192), 0, stream, emb,
                     pix_idx, m_t, lnw, lnb, An, K, P, S, S_pad, N, nseg);
  hipLaunchKernelGGL(k_contrast, dim3(B * K), dim3(128), 0, stream, An, Mn,
                     lcon, K, S, S_pad, nseg);
  hipLaunchKernelGGL(k_final, dim3(1), dim3(1), 0, stream, lpix, lcon,
                     (float*)d_out, B);
}